// BasicDeconvolutionBlock_4037269258943
// MI455X (gfx1250) — compile-verified
//
#include <hip/hip_runtime.h>
#include <hip/hip_bf16.h>
#include <stdint.h>

typedef float v2f __attribute__((ext_vector_type(2)));
typedef float v8f __attribute__((ext_vector_type(8)));

#define CIN   256
#define COUT  128
#define TM    16              // WMMA M
#define MT    4               // M-subtiles per block
#define ROWS  (TM * MT)       // 64 match rows per block
#define ASTR  260             // row-major LDS stride (floats): 1040B, 16B-aligned,
                              // fragment bank = (4m + c) % 64 -> conflict-free

__global__ __launch_bounds__(256)
void deconv_wmma_f32(const float* __restrict__ feats,
                     const float* __restrict__ weight,
                     const int*   __restrict__ gather_idx,
                     const int*   __restrict__ scatter_idx,
                     const int*   __restrict__ n_out_ptr,
                     float*       __restrict__ out,
                     int n_in, int ntiles)
{
    // A staged row-major: a_rm[r][c] at r*ASTR + c  (64*260 floats = 66.6 KB)
    __shared__ float a_rm[ROWS * ASTR];
    __shared__ int   s_idx[ROWS];
    __shared__ int   s_any;

    const int n_out = *n_out_ptr;
    const int k     = blockIdx.x / ntiles;
    const int tile  = blockIdx.x - k * ntiles;
    const int row0  = tile * ROWS;
    const int tid   = threadIdx.x;

    if (tid == 0) s_any = 0;
    __syncthreads();

    // Load the 64 scatter indices; detect all-sentinel tiles for early exit.
    if (tid < ROWS) {
        int r = row0 + tid;
        int s = (r < n_in) ? scatter_idx[k * n_in + r] : n_out;
        s_idx[tid] = s;
        if (s < n_out) atomicOr(&s_any, 1);
    }
    __syncthreads();
    if (s_any == 0) return;   // uniform across block: EXEC stays all-ones below

    // ---- Stage gathered A tile (64 x 256 fp32) via async global->LDS DMA ----
    {
        const int r  = tid >> 2;          // 0..63 : match row within tile
        const int c0 = (tid & 3) << 6;    // 0/64/128/192 : 64-column chunk
        int g = (row0 + r < n_in) ? gather_idx[k * n_in + row0 + r] : n_in;
        const bool valid = (g < n_in);    // g == n_in is the zero-pad sentinel row

        if (valid) {
            // GVS mode: mem = SADDR64(feats) + VADDR32(byte offset); LDS dest
            // from VDST VGPR. Tracked on ASYNCcnt, no VGPR data staging.
            uint32_t goff = (uint32_t)g * (CIN * 4) + (uint32_t)(c0 * 4);
            #pragma unroll
            for (int i = 0; i < 16; ++i) {
                uint32_t lds_addr =
                    (uint32_t)(size_t)&a_rm[r * ASTR + c0 + 4 * i];
                asm volatile(
                    "global_load_async_to_lds_b128 %0, %1, %2 offset:0"
                    :: "v"(lds_addr), "v"(goff + 16u * i), "s"(feats)
                    : "memory");
            }
        } else {
            const float4 z = make_float4(0.f, 0.f, 0.f, 0.f);
            #pragma unroll
            for (int i = 0; i < 16; ++i)
                *(float4*)&a_rm[r * ASTR + c0 + 4 * i] = z;
        }
    }
    // Drain this wave's async DMAs, then rendezvous with the other waves.
    asm volatile("s_wait_asynccnt 0x0" ::: "memory");
    __syncthreads();

    // ---- Per-wave 16x(4x16) output tiles: B fragment reused across 4 M-subtiles ----
    const int wave  = tid >> 5;           // 0..7 -> output column block
    const int lane  = tid & 31;
    const int m     = lane & 15;          // A row / B,D column within tile
    const int half  = lane >> 4;          // K split: 0 -> K{0,1}, 1 -> K{2,3}
    const int nbase = wave << 4;

    const float* W = weight + (size_t)k * (CIN * COUT) + nbase + m;

    v8f acc[MT];
    #pragma unroll
    for (int mt = 0; mt < MT; ++mt) acc[mt] = (v8f){};

    #pragma unroll 4
    for (int c0 = 0; c0 < CIN; c0 += 4) {
        const int ca = c0 + 2 * half;
        v2f b;
        b.x = W[(size_t)(ca    ) * COUT]; // B(ca,   nbase+m)
        b.y = W[(size_t)(ca + 1) * COUT]; // B(ca+1, nbase+m)
        #pragma unroll
        for (int mt = 0; mt < MT; ++mt) {
            // A(mt*16+m, ca..ca+1): contiguous in LDS -> single ds_load_b64
            v2f a = *(const v2f*)&a_rm[(mt * TM + m) * ASTR + ca];
            acc[mt] = __builtin_amdgcn_wmma_f32_16x16x4_f32(
                          /*neg_a=*/false, a, /*neg_b=*/false, b,
                          /*c_mod=*/(short)0, acc[mt],
                          /*reuse_a=*/false, /*reuse_b=*/false);
        }
    }

    // ---- Fused ReLU + scatter store (each output row written exactly once) ----
    #pragma unroll
    for (int mt = 0; mt < MT; ++mt) {
        #pragma unroll
        for (int v = 0; v < 8; ++v) {
            const int mrow = mt * TM + v + 8 * half;  // C/D layout: VGPR v -> M=v / v+8
            const int j = s_idx[mrow];
            if (j < n_out) {
                out[(size_t)j * COUT + nbase + m] = fmaxf(acc[mt][v], 0.f);
            }
        }
    }
}

extern "C" void kernel_launch(void* const* d_in, const int* in_sizes, int n_in_bufs,
                              void* d_out, int out_size, void* d_ws, size_t ws_size,
                              hipStream_t stream) {
    const float* feats     = (const float*)d_in[0];
    const float* weight    = (const float*)d_in[1];
    const int*   gather    = (const int*)d_in[2];
    const int*   scatter   = (const int*)d_in[3];
    const int*   n_out_ptr = (const int*)d_in[4];
    float*       out       = (float*)d_out;

    const int n_in   = in_sizes[0] / CIN;               // 100000
    const int kvol   = in_sizes[1] / (CIN * COUT);      // 8
    const int ntiles = (n_in + ROWS - 1) / ROWS;        // 1563

    dim3 grid(kvol * ntiles);
    dim3 block(256);                                    // 8 wave32 waves
    deconv_wmma_f32<<<grid, block, 0, stream>>>(feats, weight, gather, scatter,
                                                n_out_ptr, out, n_in, ntiles);
}